// SafetyGCN_26036091748418
// MI455X (gfx1250) — compile-verified
//
#include <hip/hip_runtime.h>

#define NODES_C_IN 128
#define HID 64

typedef float v2f __attribute__((ext_vector_type(2)));
typedef float v8f __attribute__((ext_vector_type(8)));

// Native no-return f32 scatter-add (STOREcnt path). Guarantees the single
// global_atomic_add_f32 instruction instead of a possible CAS-loop expansion
// of atomicrmw fadd — this matters: the edge scatter is ~102M adds per layer
// and is the roofline-dominant part of this workload.
__device__ __forceinline__ void atomic_add_f32(float* p, float v) {
    asm volatile("global_atomic_add_f32 %0, %1, off" : : "v"(p), "v"(v) : "memory");
}

// ---------------------------------------------------------------- utilities
__global__ __launch_bounds__(256) void fill_f32(float* __restrict__ p, float v, int n) {
    int i = blockIdx.x * 256 + threadIdx.x;
    if (i < n) p[i] = v;
}

__global__ __launch_bounds__(256) void count_deg(const int* __restrict__ dst,
                                                 float* __restrict__ deg, int e) {
    int i = blockIdx.x * 256 + threadIdx.x;
    if (i < e) atomic_add_f32(&deg[dst[i]], 1.0f);
}

__global__ __launch_bounds__(256) void rsqrt_deg(float* __restrict__ p, int n) {
    int i = blockIdx.x * 256 + threadIdx.x;
    if (i < n) {
        float d = p[i];
        p[i] = (d > 0.0f) ? rsqrtf(d) : 0.0f;
    }
}

// ---------------------------------------------------------------- WMMA GEMM
// C[n x 64] = A[n x K] * W[K x 64], fp32 exact via V_WMMA_F32_16X16X4_F32.
// 8 waves / block; each wave owns one 16x16 tile; block covers 32 rows x 64 cols.
__global__ __launch_bounds__(256) void gemm_wmma_f32(const float* __restrict__ A,
                                                     const float* __restrict__ W,
                                                     float* __restrict__ C,
                                                     int K, int nrows) {
    const int lane = threadIdx.x & 31;
    const int wave = threadIdx.x >> 5;
    const int rowBase = (blockIdx.x * 2 + (wave >> 2)) * 16;
    const int colBase = (wave & 3) * 16;
    if (rowBase + 16 > nrows) return;   // wave-uniform guard: EXEC all-ones around WMMA

    const int m  = lane & 15;           // row (A) / col (B,C) within tile
    const int hi = lane >> 4;           // 0: K=0,1   1: K=2,3
    const int krem = hi * 2;

    const float* arow = A + (size_t)(rowBase + m) * K;
    const float* wcol = W + colBase + m;

    v8f acc = {};
    for (int k = 0; k < K; k += 4) {
        // A 16x4 fragment: lane holds A[m][k+krem], A[m][k+krem+1]  (float2, 8B aligned)
        v2f a = *(const v2f*)(arow + k + krem);
        // B 4x16 fragment: lane holds W[k+krem][col], W[k+krem+1][col]
        v2f b;
        b.x = wcol[(size_t)(k + krem) * HID];
        b.y = wcol[(size_t)(k + krem + 1) * HID];
        acc = __builtin_amdgcn_wmma_f32_16x16x4_f32(false, a, false, b,
                                                    (short)0, acc, false, false);
    }

    // C layout: VGPR r -> row (r + 8*hi), col = lane&15
    float* crow = C + (size_t)rowBase * HID + colBase + m;
#pragma unroll
    for (int r = 0; r < 8; ++r)
        crow[(size_t)(r + 8 * hi) * HID] = acc[r];
}

// ---------------------------------------------------------------- edge aggregation
// One wave per edge; lane handles a coalesced float2 of channels.
// agg[dst] += dinv[src]*dinv[dst] * h[src]   (scatter via native f32 atomics;
// h is 25.6 MB and resident in the 192 MB L2, so gathers/atomics stay on-chip)
__global__ __launch_bounds__(256) void agg_edges(const float* __restrict__ h,
                                                 const int* __restrict__ src,
                                                 const int* __restrict__ dst,
                                                 const float* __restrict__ dinv,
                                                 float* __restrict__ agg, int e) {
    const int lane = threadIdx.x & 31;
    const int ei = blockIdx.x * 8 + (threadIdx.x >> 5);
    if (ei >= e) return;
    const int s = src[ei];
    const int d = dst[ei];
    const float nrm = dinv[s] * dinv[d];
    const int c = lane * 2;
    const float2 hv = *(const float2*)(h + (size_t)s * HID + c);
    float* outp = agg + (size_t)d * HID + c;
    atomic_add_f32(outp,     nrm * hv.x);
    atomic_add_f32(outp + 1, nrm * hv.y);
}

// ---------------------------------------------------------------- layer epilogue
// x2 = relu(agg + dinv^2 * h + b); agg is re-zeroed for the next layer.
__global__ __launch_bounds__(256) void post_layer(float* __restrict__ agg,
                                                  const float* __restrict__ h,
                                                  const float* __restrict__ dinv,
                                                  const float* __restrict__ b,
                                                  float* __restrict__ x2, int total) {
    int idx = blockIdx.x * 256 + threadIdx.x;
    if (idx >= total) return;
    const int i = idx >> 6;
    const int c = idx & 63;
    const float di = dinv[i];
    float v = agg[idx] + di * di * h[idx] + b[c];
    x2[idx] = fmaxf(v, 0.0f);
    agg[idx] = 0.0f;
}

// ---------------------------------------------------------------- final head
// logits[i] = relu(agg + dinv^2*h + b2) . Wc + bc   (one wave per node, wave32 reduce)
__global__ __launch_bounds__(256) void final_logits(const float* __restrict__ agg,
                                                    const float* __restrict__ h,
                                                    const float* __restrict__ dinv,
                                                    const float* __restrict__ b2,
                                                    const float* __restrict__ Wc,
                                                    const float* __restrict__ bc,
                                                    float* __restrict__ out, int n) {
    const int lane = threadIdx.x & 31;
    const int i = blockIdx.x * 8 + (threadIdx.x >> 5);
    if (i >= n) return;
    const float di = dinv[i];
    const float self = di * di;
    float partial = 0.0f;
#pragma unroll
    for (int t = 0; t < 2; ++t) {
        const int c = lane + t * 32;
        const size_t idx = (size_t)i * HID + c;
        float v = agg[idx] + self * h[idx] + b2[c];
        v = fmaxf(v, 0.0f);
        partial += v * Wc[c];
    }
#pragma unroll
    for (int off = 16; off > 0; off >>= 1)
        partial += __shfl_xor(partial, off, 32);
    if (lane == 0) out[i] = partial + bc[0];
}

// ---------------------------------------------------------------- launch
extern "C" void kernel_launch(void* const* d_in, const int* in_sizes, int n_in,
                              void* d_out, int out_size, void* d_ws, size_t ws_size,
                              hipStream_t stream) {
    const float* x  = (const float*)d_in[0];
    const int*   ei = (const int*)d_in[1];   // int32 on device (JAX default x64-off)
    const float* W1 = (const float*)d_in[2];
    const float* b1 = (const float*)d_in[3];
    const float* W2 = (const float*)d_in[4];
    const float* b2 = (const float*)d_in[5];
    const float* Wc = (const float*)d_in[6];
    const float* bc = (const float*)d_in[7];
    float* out = (float*)d_out;

    const int n = in_sizes[0] / NODES_C_IN;  // 100000
    const int e = in_sizes[1] / 2;           // 1600000
    const int* src = ei;
    const int* dst = ei + e;

    // workspace layout: dinv[n] | h[n*64] | agg[n*64] | x2[n*64]  (~77 MB)
    char* ws = (char*)d_ws;
    const size_t dinvBytes = (((size_t)n * 4) + 255) & ~(size_t)255;
    const size_t featBytes = (size_t)n * HID * 4;
    float* dinv = (float*)ws;
    float* h    = (float*)(ws + dinvBytes);
    float* agg  = (float*)(ws + dinvBytes + featBytes);
    float* x2   = (float*)(ws + dinvBytes + 2 * featBytes);

    const int total = n * HID;

    // degree (incl. self-loop) -> dinv = rsqrt(deg)
    fill_f32 <<<(n + 255) / 256, 256, 0, stream>>>(dinv, 1.0f, n);
    count_deg<<<(e + 255) / 256, 256, 0, stream>>>(dst, dinv, e);
    rsqrt_deg<<<(n + 255) / 256, 256, 0, stream>>>(dinv, n);
    fill_f32 <<<(total + 255) / 256, 256, 0, stream>>>(agg, 0.0f, total);

    // layer 1: h = x @ W1 ; aggregate ; relu(+bias+self-loop)
    gemm_wmma_f32<<<(n + 31) / 32, 256, 0, stream>>>(x, W1, h, NODES_C_IN, n);
    agg_edges    <<<(e + 7) / 8,   256, 0, stream>>>(h, src, dst, dinv, agg, e);
    post_layer   <<<(total + 255) / 256, 256, 0, stream>>>(agg, h, dinv, b1, x2, total);

    // layer 2: h = x2 @ W2 ; aggregate ; fused relu + classifier head
    gemm_wmma_f32<<<(n + 31) / 32, 256, 0, stream>>>(x2, W2, h, HID, n);
    agg_edges    <<<(e + 7) / 8,   256, 0, stream>>>(h, src, dst, dinv, agg, e);
    final_logits <<<(n + 7) / 8,   256, 0, stream>>>(agg, h, dinv, b2, Wc, bc, out, n);
}